// NoiseContrastiveEstimation_62070867361884
// MI455X (gfx1250) — compile-verified
//
#include <hip/hip_runtime.h>
#include <hip/hip_bf16.h>
#include <math.h>

typedef __attribute__((ext_vector_type(2))) float v2f;
typedef __attribute__((ext_vector_type(8))) float v8f;

#define VOCAB   1000000
#define EMB     128
#define NEG     512
#define BATCH   16384
#define BM      128
#define BN      64
#define LDT     130   // padded LDS row stride (floats): 130 % 64 == 2 -> conflict-free b64 frag loads

// log(vocab+1) reciprocal, computed on device each time (cheap, deterministic)
__device__ __forceinline__ float bias_adjust(int id, const float* __restrict__ biases) {
    // P(c) = (log(c+2)-log(c+1)) / log(VOCAB+1);  E[count] = -expm1(NEG*log1p(-P))
    float c  = (float)id;
    float p  = (logf(c + 2.0f) - logf(c + 1.0f)) / logf((float)VOCAB + 1.0f);
    float ec = -expm1f((float)NEG * log1pf(-p));
    return biases[id] - logf(ec);
}

__device__ __forceinline__ float sigmoid_ce(float l, float z) {
    // max(l,0) - l*z + log1p(exp(-|l|))
    return fmaxf(l, 0.0f) - l * z + log1pf(expf(-fabsf(l)));
}

// ---------------------------------------------------------------------------
// Kernel 1: true-class logits. One wave32 per batch row: 128-elem dot product
// via float4 per lane + shuffle reduction. Memory bound (gathered rows hit L2).
// ---------------------------------------------------------------------------
__global__ __launch_bounds__(256) void nce_true_kernel(
    const float* __restrict__ pred,     // [BATCH, EMB]
    const float* __restrict__ weights,  // [VOCAB, EMB]
    const float* __restrict__ biases,   // [VOCAB]
    const int*   __restrict__ labels,   // [BATCH]
    float* __restrict__ acc)
{
    __shared__ float lAcc;
    if (threadIdx.x == 0) lAcc = 0.0f;
    __syncthreads();

    const int wv   = threadIdx.x >> 5;
    const int lane = threadIdx.x & 31;
    const int b    = blockIdx.x * 8 + wv;   // 2048 blocks * 8 waves = 16384 rows

    const int lab = labels[b];
    const float4 p = ((const float4*)(pred    + (size_t)b   * EMB))[lane];
    const float4 w = ((const float4*)(weights + (size_t)lab * EMB))[lane];
    float d = p.x * w.x + p.y * w.y + p.z * w.z + p.w * w.w;

    #pragma unroll
    for (int off = 16; off > 0; off >>= 1)
        d += __shfl_down(d, off, 32);

    if (lane == 0) {
        float l  = d + bias_adjust(lab, biases);
        float ce = sigmoid_ce(l, 1.0f);          // num_true == 1 -> z = 1
        atomicAdd(&lAcc, ce);
    }
    __syncthreads();
    if (threadIdx.x == 0) atomicAdd(acc, lAcc);
}

// ---------------------------------------------------------------------------
// Kernel 2: sampled logits GEMM (pred @ Ws^T) via V_WMMA_F32_16X16X4_F32 with
// fused bias/log-expected-count + sigmoid-CE epilogue and block reduction.
// Block = 256 threads (8 waves). Tile BM=128 x BN=64, K=128 fully resident.
// Each wave computes a 16x64 output strip -> 128 WMMAs per wave.
// ---------------------------------------------------------------------------
__global__ __launch_bounds__(256) void nce_sampled_kernel(
    const float* __restrict__ pred,     // [BATCH, EMB]
    const float* __restrict__ weights,  // [VOCAB, EMB]
    const float* __restrict__ biases,   // [VOCAB]
    const int*   __restrict__ sampled,  // [NEG]
    float* __restrict__ acc)
{
    __shared__ float lA[BM * LDT];      // pred tile, padded stride
    __shared__ float lB[BN * LDT];      // gathered sampled-weight tile
    __shared__ float lBias[BN];         // biases[s] - log(E[count](s))
    __shared__ float lAcc;

    const int t     = threadIdx.x;
    const int mBase = blockIdx.x * BM;
    const int nBase = blockIdx.y * BN;

    if (t == 0) lAcc = 0.0f;

    // ---- stage pred tile: 128 rows x 32 float4 = 4096 float4, 16 per thread
    {
        const float4* gp = (const float4*)(pred + (size_t)mBase * EMB);
        #pragma unroll
        for (int i = 0; i < 16; ++i) {
            int id = t + 256 * i;
            int r  = id >> 5;
            int c4 = id & 31;
            float4 v = gp[(size_t)r * 32 + c4];
            float* dst = &lA[r * LDT + c4 * 4];
            dst[0] = v.x; dst[1] = v.y; dst[2] = v.z; dst[3] = v.w;
        }
    }
    // ---- gather Ws tile: 64 rows x 32 float4 = 2048 float4, 8 per thread
    {
        #pragma unroll
        for (int i = 0; i < 8; ++i) {
            int id = t + 256 * i;
            int s  = id >> 5;
            int c4 = id & 31;
            int row = sampled[nBase + s];
            float4 v = ((const float4*)(weights + (size_t)row * EMB))[c4];
            float* dst = &lB[s * LDT + c4 * 4];
            dst[0] = v.x; dst[1] = v.y; dst[2] = v.z; dst[3] = v.w;
        }
    }
    if (t < BN) {
        lBias[t] = bias_adjust(sampled[nBase + t], biases);
    }
    __syncthreads();

    const int lane = t & 31;
    const int wv   = t >> 5;        // wave 0..7 -> M strip
    const int ln   = lane & 15;     // M (for A) / N (for B,C,D)
    const int hi   = lane >> 4;     // K-pair selector (0 -> K{0,1}, 1 -> K{2,3})

    // ISA fragment layout: A lane ln holds row M=ln, dwords K = 2*hi + {0,1}
    const float* aPtr = &lA[(wv * 16 + ln) * LDT + 2 * hi];
    const float* bPtr0 = &lB[( 0 + ln) * LDT + 2 * hi];
    const float* bPtr1 = &lB[(16 + ln) * LDT + 2 * hi];
    const float* bPtr2 = &lB[(32 + ln) * LDT + 2 * hi];
    const float* bPtr3 = &lB[(48 + ln) * LDT + 2 * hi];

    v8f c0 = {}, c1 = {}, c2 = {}, c3 = {};

    #pragma unroll
    for (int k = 0; k < EMB; k += 4) {
        v2f a  = *(const v2f*)(aPtr  + k);
        v2f b0 = *(const v2f*)(bPtr0 + k);
        v2f b1 = *(const v2f*)(bPtr1 + k);
        v2f b2 = *(const v2f*)(bPtr2 + k);
        v2f b3 = *(const v2f*)(bPtr3 + k);
        c0 = __builtin_amdgcn_wmma_f32_16x16x4_f32(false, a, false, b0, (short)0, c0, false, false);
        c1 = __builtin_amdgcn_wmma_f32_16x16x4_f32(false, a, false, b1, (short)0, c1, false, false);
        c2 = __builtin_amdgcn_wmma_f32_16x16x4_f32(false, a, false, b2, (short)0, c2, false, false);
        c3 = __builtin_amdgcn_wmma_f32_16x16x4_f32(false, a, false, b3, (short)0, c3, false, false);
    }

    // ---- epilogue: D element (vgpr r, lane L) = logits[M = r + 8*hi][N = ln]
    float partial = 0.0f;
    {
        float badj0 = lBias[ 0 + ln];
        float badj1 = lBias[16 + ln];
        float badj2 = lBias[32 + ln];
        float badj3 = lBias[48 + ln];
        #pragma unroll
        for (int r = 0; r < 8; ++r) {
            partial += sigmoid_ce(c0[r] + badj0, 0.0f);
            partial += sigmoid_ce(c1[r] + badj1, 0.0f);
            partial += sigmoid_ce(c2[r] + badj2, 0.0f);
            partial += sigmoid_ce(c3[r] + badj3, 0.0f);
        }
    }

    atomicAdd(&lAcc, partial);      // ds_add_f32
    __syncthreads();
    if (t == 0) atomicAdd(acc, lAcc);
}

// ---------------------------------------------------------------------------
// Kernel 3: finalize -> mean over batch
// ---------------------------------------------------------------------------
__global__ void nce_finalize_kernel(const float* __restrict__ acc, float* __restrict__ out)
{
    if (threadIdx.x == 0 && blockIdx.x == 0)
        out[0] = acc[0] * (1.0f / (float)BATCH);
}

extern "C" void kernel_launch(void* const* d_in, const int* in_sizes, int n_in,
                              void* d_out, int out_size, void* d_ws, size_t ws_size,
                              hipStream_t stream) {
    const float* pred    = (const float*)d_in[0];   // [16384,128]
    const float* weights = (const float*)d_in[1];   // [1000000,128]
    const float* biases  = (const float*)d_in[2];   // [1000000]
    const int*   labels  = (const int*)d_in[3];     // [16384,1]
    const int*   sampled = (const int*)d_in[4];     // [512]
    float* out = (float*)d_out;
    float* acc = (float*)d_ws;                      // 4-byte scalar accumulator

    hipMemsetAsync(acc, 0, sizeof(float), stream);

    nce_true_kernel<<<BATCH / 8, 256, 0, stream>>>(pred, weights, biases, labels, acc);

    dim3 grid(BATCH / BM, NEG / BN);                // 128 x 8
    nce_sampled_kernel<<<grid, 256, 0, stream>>>(pred, weights, biases, sampled, acc);

    nce_finalize_kernel<<<1, 1, 0, stream>>>(acc, out);
}